// RACE_65618510348699
// MI455X (gfx1250) — compile-verified
//
#include <hip/hip_runtime.h>
#include <math.h>

// MI455X / gfx1250, wave32. Radial-only reduction of the reference GNN
// (m>0 channels of h are dead; Y[:,0]==1 => forces flow only through r).
// Edge MLP (8->64->64->32) fwd+bwd done with V_WMMA_F32_16X16X4_F32 on
// 16-edge tiles; scatters via global_atomic_add_f32.

#define CUTOFF_F 6.0f
#define AVGN_F   40.0f
#define PI_F     3.14159265358979f
#define SQRT2_F  1.41421356237f

typedef float v2f __attribute__((ext_vector_type(2)));
typedef float v8f __attribute__((ext_vector_type(8)));

__device__ __forceinline__ v8f wmma4(v2f a, v2f b, v8f c) {
  // D = A(16x4 f32) * B(4x16 f32) + C(16x16 f32)
  return __builtin_amdgcn_wmma_f32_16x16x4_f32(false, a, false, b, (short)0, c, false, false);
}
__device__ __forceinline__ v8f zero8() { v8f z = {0.f,0.f,0.f,0.f,0.f,0.f,0.f,0.f}; return z; }

__device__ __forceinline__ float frcp(float x)   { return __builtin_amdgcn_rcpf(x); }   // v_rcp_f32
__device__ __forceinline__ float sigf(float x)   { return frcp(1.f + __expf(-x)); }
__device__ __forceinline__ float siluf(float x)  { return x * sigf(x); }
__device__ __forceinline__ float dsiluf(float x) { float s = sigf(x); return s*(1.f + x*(1.f - s)); }
__device__ __forceinline__ float envf(float r)   { float r2=r*r; return (r<1.f)?(1.f-6.f*r2+8.f*r2*r-3.f*r2*r2):0.f; }
__device__ __forceinline__ float denvf(float r)  { float r2=r*r; return (r<1.f)?(-12.f*r+24.f*r2-12.f*r2*r):0.f; }
#define LDS_FENCE() asm volatile("s_wait_dscnt 0" ::: "memory")

// ---------------- utility ----------------
__global__ void k_zero(float* __restrict__ p, int n) {
  int i = blockIdx.x*blockDim.x + threadIdx.x;
  if (i < n) p[i] = 0.f;
}

// ---------------- per-edge geometry: rdir = {u/CUTOFF (3), r} ----------------
__global__ void k_geom(int E, int EperG, const float* __restrict__ R, const float* __restrict__ cell,
                       const float* __restrict__ Sij, const int* __restrict__ eidx,
                       float* __restrict__ rdir) {
  int e = blockIdx.x*blockDim.x + threadIdx.x;
  if (e >= E) return;
  int i = eidx[e], j = eidx[E+e];
  int g = e / EperG;
  const float* cl = cell + g*9;
  float sx = Sij[e*3+0], sy = Sij[e*3+1], sz = Sij[e*3+2];
  float rn[3];
  #pragma unroll
  for (int k=0;k<3;k++) {
    float sh = sx*cl[k] + sy*cl[3+k] + sz*cl[6+k];
    rn[k] = (R[j*3+k] - R[i*3+k] + sh) * (1.f/CUTOFF_F);
  }
  float r = sqrtf(rn[0]*rn[0]+rn[1]*rn[1]+rn[2]*rn[2]) + 1e-12f;
  float s = frcp(r*CUTOFF_F);
  rdir[e*4+0] = rn[0]*s;
  rdir[e*4+1] = rn[1]*s;
  rdir[e*4+2] = rn[2]*s;
  rdir[e*4+3] = r;
}

// ---------------- node embedding: h0, xs = silu(h0 @ Wx) ----------------
__global__ void k_embed(int N, const int* __restrict__ species, const float* __restrict__ We,
                        const float* __restrict__ Wx, float* __restrict__ h0, float* __restrict__ xs) {
  int id = blockIdx.x*blockDim.x + threadIdx.x;
  if (id >= N*32) return;
  int n = id >> 5, c = id & 31;
  int sp = species[n];
  h0[id] = We[sp*32+c];
  float acc = 0.f;
  for (int k=0;k<32;k++) acc += We[sp*32+k]*Wx[k*32+c];
  xs[id] = siluf(acc);
}

// ---------------- forward edge kernel (per layer): radial MLP + message scatter ----------------
__global__ __launch_bounds__(128) void k_edge_fwd(
    int l, int E, int N,
    const float* __restrict__ rdir, const int* __restrict__ eidx,
    const float* __restrict__ hprev,
    const float* __restrict__ Wr1, const float* __restrict__ Wr2, const float* __restrict__ Wr3,
    float* __restrict__ A0) {
  __shared__ float sW1[8*64];
  __shared__ float sW2[64*64];
  __shared__ float sW3[64*32];     // Wr3 columns c*4 only (l=0 of LOF)
  __shared__ float sZ[4][16*64];   // per-wave staging
  __shared__ int   sIJ[4][32];
  const float* wr1 = Wr1 + (size_t)l*8*64;
  const float* wr2 = Wr2 + (size_t)l*64*64;
  const float* wr3 = Wr3 + (size_t)l*64*128;
  for (int t=threadIdx.x; t<8*64;  t+=128) sW1[t] = wr1[t];
  for (int t=threadIdx.x; t<64*64; t+=128) sW2[t] = wr2[t];
  for (int t=threadIdx.x; t<64*32; t+=128) { int f=t>>5, c=t&31; sW3[t] = wr3[f*128 + c*4]; }
  __syncthreads();
  const int wid = threadIdx.x>>5, lane = threadIdx.x&31, lo = lane&15, hi = lane>>4;
  float* zb = sZ[wid];
  int* si = sIJ[wid]; int* sj = sIJ[wid] + 16;
  // Wr1 B-fragments live in registers (16 VGPRs)
  v2f b1[2][4];
  #pragma unroll
  for (int kb=0;kb<2;kb++)
    #pragma unroll
    for (int nt=0;nt<4;nt++) {
      int k = kb*4 + hi*2, n = nt*16 + lo;
      v2f b; b.x = sW1[k*64+n]; b.y = sW1[(k+1)*64+n]; b1[kb][nt] = b;
    }
  int ntiles = E >> 4;
  for (int t = blockIdx.x*4 + wid; t < ntiles; t += gridDim.x*4) {
    int eb = t << 4;
    {   // prefetch next tile's gather sources
      int t2 = t + gridDim.x*4;
      if (t2 < ntiles) {
        __builtin_prefetch(&rdir[(((size_t)t2<<4)+lo)*4], 0, 1);
        __builtin_prefetch(&eidx[((size_t)t2<<4)+lo], 0, 1);
      }
    }
    if (lane < 16) { si[lane] = eidx[eb+lane]; sj[lane] = eidx[E+eb+lane]; }
    // ef A-fragments (recompute radial basis from r):  ef_n = (sqrt2*env/r) * sin(n*pi*r)
    float r = rdir[(eb+lo)*4+3];
    float scale = SQRT2_F * envf(r) * frcp(r);
    v2f a0[2];
    #pragma unroll
    for (int kb=0;kb<2;kb++) {
      int k = kb*4 + hi*2;
      v2f a;
      a.x = scale * __sinf((float)(k+1)*PI_F*r);
      a.y = scale * __sinf((float)(k+2)*PI_F*r);
      a0[kb] = a;
    }
    // z1 = silu(ef @ Wr1) : 16x64
    #pragma unroll
    for (int nt=0;nt<4;nt++) {
      v8f acc = zero8();
      acc = wmma4(a0[0], b1[0][nt], acc);
      acc = wmma4(a0[1], b1[1][nt], acc);
      #pragma unroll
      for (int v=0;v<8;v++) zb[(v+8*hi)*64 + nt*16+lo] = siluf(acc[v]);
    }
    LDS_FENCE();
    // z2 = silu(z1 @ Wr2) : 16x64
    v8f z2[4];
    #pragma unroll
    for (int nt=0;nt<4;nt++) {
      v8f acc = zero8();
      #pragma unroll
      for (int kb=0;kb<16;kb++) {
        int k = kb*4 + hi*2;
        v2f a = *(const v2f*)&zb[lo*64 + k];
        v2f b; b.x = sW2[k*64 + nt*16+lo]; b.y = sW2[(k+1)*64 + nt*16+lo];
        acc = wmma4(a, b, acc);
      }
      z2[nt] = acc;
    }
    #pragma unroll
    for (int nt=0;nt<4;nt++)
      #pragma unroll
      for (int v=0;v<8;v++) zb[(v+8*hi)*64 + nt*16+lo] = siluf(z2[nt][v]);
    LDS_FENCE();
    // w0 = z2 @ W3c : 16x32 ; msg = w0 * hprev[j]; scatter into A0[i]
    #pragma unroll
    for (int nt=0;nt<2;nt++) {
      v8f acc = zero8();
      #pragma unroll
      for (int kb=0;kb<16;kb++) {
        int k = kb*4 + hi*2;
        v2f a = *(const v2f*)&zb[lo*64 + k];
        v2f b; b.x = sW3[k*32 + nt*16+lo]; b.y = sW3[(k+1)*32 + nt*16+lo];
        acc = wmma4(a, b, acc);
      }
      int c = nt*16 + lo;
      #pragma unroll
      for (int v=0;v<8;v++) {
        int m = v + 8*hi;
        int i = si[m], j = sj[m];
        float msg = acc[v] * hprev[j*32+c];
        atomicAdd(&A0[i*32+c], msg);
      }
    }
    LDS_FENCE();
  }
}

// ---------------- forward node kernel (per layer) ----------------
__global__ __launch_bounds__(256) void k_node_fwd(
    int l, int N,
    const float* __restrict__ A0, const float* __restrict__ xs,
    const float* __restrict__ hprev, float* __restrict__ hout,
    const float* __restrict__ Wp, const float* __restrict__ Wsc,
    const float* __restrict__ Wro1, const float* __restrict__ Wro2,
    float* __restrict__ outs) {
  __shared__ float sHp[8][32], sHv[8][32], sHc[8][32];
  int ln = threadIdx.x>>5, c = threadIdx.x&31;
  int n = blockIdx.x*8 + ln;
  const float* wp  = Wp  + (size_t)(l*4)*32*32;  // Wp[l,0]
  const float* wsc = Wsc + (size_t)l*32*32;
  if (n < N) {
    sHp[ln][c] = A0[n*32+c]*(1.f/AVGN_F)*xs[n*32+c];
    sHv[ln][c] = hprev[n*32+c];
  } else { sHp[ln][c]=0.f; sHv[ln][c]=0.f; }
  __syncthreads();
  float acc = 0.f;
  for (int k=0;k<32;k++) acc += sHp[ln][k]*wp[k*32+c] + sHv[ln][k]*wsc[k*32+c];
  sHc[ln][c] = acc;
  if (n < N) hout[n*32+c] = acc;
  __syncthreads();
  const float* w1 = Wro1 + (size_t)l*32*64; const float* w2 = Wro2 + (size_t)l*64;
  int o = threadIdx.x & 63;
  for (int s=0;s<2;s++) {
    int ln2 = (threadIdx.x>>6) + s*4;
    int n2 = blockIdx.x*8 + ln2;
    if (n2 < N) {
      float pre = 0.f;
      for (int k=0;k<32;k++) pre += sHc[ln2][k]*w1[k*64+o];
      atomicAdd(&outs[n2*3+l], siluf(pre)*w2[o]);
    }
  }
}

// ---------------- energy / var / g_outs ----------------
__global__ void k_energy(int N, float invNG, const float* __restrict__ outs, const int* __restrict__ batch,
                         float* __restrict__ energy, float* __restrict__ var, float* __restrict__ gouts) {
  int n = blockIdx.x*blockDim.x + threadIdx.x;
  if (n >= N) return;
  float ne = 0.f;
  #pragma unroll
  for (int l=0;l<3;l++) { float o = outs[n*3+l]; ne += siluf(o); gouts[n*3+l] = dsiluf(o); }
  int g = batch[n];
  atomicAdd(&energy[g], ne);
  atomicAdd(&var[g], invNG);
}

// ---------------- backward node kernel (per layer) ----------------
__global__ __launch_bounds__(256) void k_node_bwd(
    int l, int N,
    const float* __restrict__ hcur,     // hcol[l+1]
    const float* __restrict__ xs,
    const float* __restrict__ gouts,
    float* __restrict__ gH_l,           // in: edge-scattered partial; out: total
    const float* __restrict__ gH_next,  // total grad of layer l+1 (null if l==2)
    const float* __restrict__ Wp, const float* __restrict__ Wsc_next,
    const float* __restrict__ Wro1, const float* __restrict__ Wro2,
    float* __restrict__ gA) {
  __shared__ float sH[8][32], sT[8][64], sG[8][32];
  int ln = threadIdx.x>>5, c = threadIdx.x&31;
  int n = blockIdx.x*8 + ln;
  sH[ln][c] = (n < N) ? hcur[n*32+c] : 0.f;
  __syncthreads();
  const float* w1 = Wro1 + (size_t)l*32*64; const float* w2 = Wro2 + (size_t)l*64;
  int o = threadIdx.x & 63;
  for (int s=0;s<2;s++) {
    int ln2 = (threadIdx.x>>6) + s*4;
    int n2 = blockIdx.x*8 + ln2;
    float pre = 0.f;
    for (int k=0;k<32;k++) pre += sH[ln2][k]*w1[k*64+o];
    float go = (n2 < N) ? gouts[n2*3+l] : 0.f;
    sT[ln2][o] = go*w2[o]*dsiluf(pre);
  }
  __syncthreads();
  float g = (n < N) ? gH_l[n*32+c] : 0.f;
  for (int o2=0;o2<64;o2++) g += sT[ln][o2]*w1[c*64+o2];
  if (gH_next && n < N) {
    for (int d=0;d<32;d++) g += gH_next[n*32+d]*Wsc_next[c*32+d];
  }
  sG[ln][c] = g;
  if (n < N) gH_l[n*32+c] = g;
  __syncthreads();
  const float* wp = Wp + (size_t)(l*4)*32*32;
  float ga = 0.f;
  for (int d=0;d<32;d++) ga += sG[ln][d]*wp[c*32+d];
  if (n < N) gA[n*32+c] = ga*xs[n*32+c];
}

// ---------------- backward edge kernel (per layer) ----------------
__global__ __launch_bounds__(128) void k_edge_bwd(
    int l, int E, int N,
    const float* __restrict__ rdir, const int* __restrict__ eidx,
    const float* __restrict__ hprev, const float* __restrict__ gA,
    float* __restrict__ gHprev,   // null for l==0
    const float* __restrict__ Wr1, const float* __restrict__ Wr2, const float* __restrict__ Wr3,
    float* __restrict__ F) {
  __shared__ float sW1[16*64];   // rows 8..15 zero-padded (transposed reads)
  __shared__ float sW2[64*64];
  __shared__ float sW3[64*32];
  __shared__ float sZ[4][16*64];
  __shared__ float sG2[4][16*64];
  __shared__ int   sIJ[4][32];
  const float* wr1 = Wr1 + (size_t)l*8*64;
  const float* wr2 = Wr2 + (size_t)l*64*64;
  const float* wr3 = Wr3 + (size_t)l*64*128;
  for (int t=threadIdx.x; t<16*64; t+=128) sW1[t] = (t < 8*64) ? wr1[t] : 0.f;
  for (int t=threadIdx.x; t<64*64; t+=128) sW2[t] = wr2[t];
  for (int t=threadIdx.x; t<64*32; t+=128) { int f=t>>5, c=t&31; sW3[t] = wr3[f*128 + c*4]; }
  __syncthreads();
  const int wid = threadIdx.x>>5, lane = threadIdx.x&31, lo = lane&15, hi = lane>>4;
  float* zb = sZ[wid]; float* gb = sG2[wid];
  int* si = sIJ[wid]; int* sj = sIJ[wid] + 16;
  int ntiles = E >> 4;
  for (int t = blockIdx.x*4 + wid; t < ntiles; t += gridDim.x*4) {
    int eb = t << 4;
    {   // prefetch next tile's gather sources
      int t2 = t + gridDim.x*4;
      if (t2 < ntiles) {
        __builtin_prefetch(&rdir[(((size_t)t2<<4)+lo)*4], 0, 1);
        __builtin_prefetch(&eidx[((size_t)t2<<4)+lo], 0, 1);
      }
    }
    if (lane < 16) { si[lane] = eidx[eb+lane]; sj[lane] = eidx[E+eb+lane]; }
    float r = rdir[(eb+lo)*4+3];
    float scale = SQRT2_F * envf(r) * frcp(r);
    v2f a0[2];
    #pragma unroll
    for (int kb=0;kb<2;kb++) {
      int k = kb*4 + hi*2;
      v2f a;
      a.x = scale * __sinf((float)(k+1)*PI_F*r);
      a.y = scale * __sinf((float)(k+2)*PI_F*r);
      a0[kb] = a;
    }
    // recompute z1 (keep silu' in regs, silu value to LDS)
    v8f dz1[4];
    #pragma unroll
    for (int nt=0;nt<4;nt++) {
      v8f acc = zero8();
      #pragma unroll
      for (int kb=0;kb<2;kb++) {
        int k = kb*4 + hi*2;
        v2f b; b.x = sW1[k*64 + nt*16+lo]; b.y = sW1[(k+1)*64 + nt*16+lo];
        acc = wmma4(a0[kb], b, acc);
      }
      #pragma unroll
      for (int v=0;v<8;v++) { zb[(v+8*hi)*64 + nt*16+lo] = siluf(acc[v]); dz1[nt][v] = dsiluf(acc[v]); }
    }
    LDS_FENCE();
    // recompute z2
    v8f dz2[4];
    #pragma unroll
    for (int nt=0;nt<4;nt++) {
      v8f acc = zero8();
      #pragma unroll
      for (int kb=0;kb<16;kb++) {
        int k = kb*4 + hi*2;
        v2f a = *(const v2f*)&zb[lo*64 + k];
        v2f b; b.x = sW2[k*64 + nt*16+lo]; b.y = sW2[(k+1)*64 + nt*16+lo];
        acc = wmma4(a, b, acc);
      }
      #pragma unroll
      for (int v=0;v<8;v++) { gb[(v+8*hi)*64 + nt*16+lo] = siluf(acc[v]); dz2[nt][v] = dsiluf(acc[v]); }
    }
    LDS_FENCE();
    // recompute w0; g_w0 = gA[i]/40 * hprev[j]; scatter g_hprev += gA[i]/40 * w0
    #pragma unroll
    for (int nt=0;nt<2;nt++) {
      v8f acc = zero8();
      #pragma unroll
      for (int kb=0;kb<16;kb++) {
        int k = kb*4 + hi*2;
        v2f a = *(const v2f*)&gb[lo*64 + k];
        v2f b; b.x = sW3[k*32 + nt*16+lo]; b.y = sW3[(k+1)*32 + nt*16+lo];
        acc = wmma4(a, b, acc);
      }
      int c = nt*16 + lo;
      #pragma unroll
      for (int v=0;v<8;v++) {
        int m = v + 8*hi;
        int i = si[m], j = sj[m];
        float ga = gA[i*32+c]*(1.f/AVGN_F);
        float hp = hprev[j*32+c];
        if (gHprev) atomicAdd(&gHprev[j*32+c], ga*acc[v]);
        zb[m*32+c] = ga*hp;   // g_w0 in [16e x 32c]
      }
    }
    LDS_FENCE();
    // g_z2pre = (g_w0 @ W3c^T) * silu'(z2pre)
    #pragma unroll
    for (int nt=0;nt<4;nt++) {
      v8f acc = zero8();
      #pragma unroll
      for (int kb=0;kb<8;kb++) {
        v2f a = *(const v2f*)&zb[lo*32 + kb*4 + hi*2];
        v2f b = *(const v2f*)&sW3[(nt*16+lo)*32 + kb*4 + hi*2];
        acc = wmma4(a, b, acc);
      }
      #pragma unroll
      for (int v=0;v<8;v++) gb[(v+8*hi)*64 + nt*16+lo] = acc[v]*dz2[nt][v];
    }
    LDS_FENCE();
    // g_z1pre = (g_z2pre @ Wr2^T) * silu'(z1pre)
    v2f az[16];
    #pragma unroll
    for (int kb=0;kb<16;kb++) az[kb] = *(const v2f*)&gb[lo*64 + kb*4 + hi*2];
    #pragma unroll
    for (int nt=0;nt<4;nt++) {
      v8f acc = zero8();
      #pragma unroll
      for (int kb=0;kb<16;kb++) {
        v2f b = *(const v2f*)&sW2[(nt*16+lo)*64 + kb*4 + hi*2];
        acc = wmma4(az[kb], b, acc);
      }
      #pragma unroll
      for (int v=0;v<8;v++) zb[(v+8*hi)*64 + nt*16+lo] = acc[v]*dz1[nt][v];
    }
    LDS_FENCE();
    // g_ef = g_z1pre @ Wr1^T  (cols 8..15 hit zero padding)
    {
      v8f acc = zero8();
      #pragma unroll
      for (int kb=0;kb<16;kb++) {
        v2f a = *(const v2f*)&zb[lo*64 + kb*4 + hi*2];
        v2f b = *(const v2f*)&sW1[lo*64 + kb*4 + hi*2];
        acc = wmma4(a, b, acc);
      }
      #pragma unroll
      for (int v=0;v<8;v++) if (lo < 8) zb[(v+8*hi)*8 + lo] = acc[v];
    }
    LDS_FENCE();
    // g_r = sum_b g_ef[b] * d(ef_b)/dr ; forces += / -=
    if (lane < 16) {
      int m = lane, e = eb + m;
      float rr = rdir[e*4+3];
      float rinv = frcp(rr);
      float ev = envf(rr), dev = denvf(rr);
      float A = SQRT2_F*ev*rinv;                      // coeff of n*pi*cos(n*pi*r)
      float B = SQRT2_F*(dev*rinv - ev*rinv*rinv);    // coeff of sin(n*pi*r)
      float gr = 0.f;
      #pragma unroll
      for (int b=0;b<8;b++) {
        float npi = (float)(b+1)*PI_F;
        gr += zb[m*8+b]*(A*npi*__cosf(npi*rr) + B*__sinf(npi*rr));
      }
      float dkx = rdir[e*4+0], dky = rdir[e*4+1], dkz = rdir[e*4+2];
      int i = si[m], j = sj[m];
      atomicAdd(&F[i*3+0],  gr*dkx); atomicAdd(&F[i*3+1],  gr*dky); atomicAdd(&F[i*3+2],  gr*dkz);
      atomicAdd(&F[j*3+0], -gr*dkx); atomicAdd(&F[j*3+1], -gr*dky); atomicAdd(&F[j*3+2], -gr*dkz);
    }
    LDS_FENCE();
  }
}

// ---------------- host orchestration ----------------
extern "C" void kernel_launch(void* const* d_in, const int* in_sizes, int n_in,
                              void* d_out, int out_size, void* d_ws, size_t ws_size,
                              hipStream_t stream) {
  const float* R    = (const float*)d_in[0];
  const float* cell = (const float*)d_in[1];
  const float* Sij  = (const float*)d_in[2];
  const float* We   = (const float*)d_in[3];
  const float* Wx   = (const float*)d_in[4];
  const float* Wr1  = (const float*)d_in[5];
  const float* Wr2  = (const float*)d_in[6];
  const float* Wr3  = (const float*)d_in[7];
  const float* Wsc  = (const float*)d_in[8];
  const float* Wp   = (const float*)d_in[9];
  const float* Wro1 = (const float*)d_in[10];
  const float* Wro2 = (const float*)d_in[11];
  const int* eidx    = (const int*)d_in[12];
  const int* species = (const int*)d_in[13];
  const int* batch   = (const int*)d_in[14];
  int N = in_sizes[13];
  int E = in_sizes[2] / 3;
  int G = in_sizes[1] / 9;
  (void)n_in; (void)out_size; (void)ws_size;

  float* out = (float*)d_out;
  float* energy = out;
  float* var    = out + G;
  float* F      = out + 2*G;

  float* w = (float*)d_ws;
  size_t off = 0;
  float* rdir  = w + off; off += (size_t)E*4;
  float* hcol  = w + off; off += (size_t)4*N*32;   // h0 + hcol after each layer
  float* xs    = w + off; off += (size_t)N*32;
  float* A0    = w + off; off += (size_t)N*32;
  float* outs  = w + off; off += (size_t)N*3;
  float* gouts = w + off; off += (size_t)N*3;
  float* gH    = w + off; off += (size_t)3*N*32;
  float* gA    = w + off; off += (size_t)N*32;

  auto zero = [&](float* p, size_t n) {
    k_zero<<<(int)((n+255)/256), 256, 0, stream>>>(p, (int)n);
  };
  zero(out, (size_t)2*G + (size_t)N*3);
  zero(outs, (size_t)N*3);
  zero(gH, (size_t)3*N*32);

  k_geom<<<(E+255)/256, 256, 0, stream>>>(E, E/G, R, cell, Sij, eidx, rdir);
  k_embed<<<(N*32+255)/256, 256, 0, stream>>>(N, species, We, Wx, hcol, xs);

  for (int l=0; l<3; l++) {
    zero(A0, (size_t)N*32);
    k_edge_fwd<<<320, 128, 0, stream>>>(l, E, N, rdir, eidx, hcol + (size_t)l*N*32,
                                        Wr1, Wr2, Wr3, A0);
    k_node_fwd<<<(N+7)/8, 256, 0, stream>>>(l, N, A0, xs, hcol + (size_t)l*N*32,
                                            hcol + (size_t)(l+1)*N*32,
                                            Wp, Wsc, Wro1, Wro2, outs);
  }
  k_energy<<<(N+255)/256, 256, 0, stream>>>(N, 1.f/(float)(N/G), outs, batch, energy, var, gouts);

  for (int l=2; l>=0; l--) {
    k_node_bwd<<<(N+7)/8, 256, 0, stream>>>(l, N, hcol + (size_t)(l+1)*N*32, xs, gouts,
                                            gH + (size_t)l*N*32,
                                            (l<2) ? gH + (size_t)(l+1)*N*32 : nullptr,
                                            Wp, (l<2) ? Wsc + (size_t)(l+1)*32*32 : Wsc,
                                            Wro1, Wro2, gA);
    k_edge_bwd<<<320, 128, 0, stream>>>(l, E, N, rdir, eidx, hcol + (size_t)l*N*32, gA,
                                        (l>0) ? gH + (size_t)(l-1)*N*32 : nullptr,
                                        Wr1, Wr2, Wr3, F);
  }
}